// ModulatedDeformConv_40441412059305
// MI455X (gfx1250) — compile-verified
//
#include <hip/hip_runtime.h>
#include <math.h>

// Problem constants (fixed by the reference harness)
constexpr int B_  = 4;
constexpr int C_  = 64;
constexpr int O_  = 64;
constexpr int H_  = 128;
constexpr int W_  = 128;
constexpr int KK_ = 9;            // 3x3
constexpr int HW_ = H_ * W_;
constexpr int KDIM = C_ * KK_;    // 576 (k index = c*9 + kk, matches dcn_w flat layout)

typedef float v2f __attribute__((ext_vector_type(2)));
typedef float v8f __attribute__((ext_vector_type(8)));

// ---------------------------------------------------------------------------
// Kernel 1: 3x3 offset/mask conv (27 outputs), sigmoid on mask channels.
// One thread per output pixel; 27*64*9 weights staged in LDS (62208 B).
// ---------------------------------------------------------------------------
__global__ __launch_bounds__(256)
void dcn_offset_kernel(const float* __restrict__ x,
                       const float* __restrict__ ow,   // [27][64][3][3]
                       const float* __restrict__ ob,   // [27]
                       float* __restrict__ dyA,        // [B][9][H][W]
                       float* __restrict__ dxA,
                       float* __restrict__ mA)
{
    __shared__ float s_w[27 * KDIM];                   // 27*576 floats
    for (int i = threadIdx.x; i < 27 * KDIM; i += 256) s_w[i] = ow[i];
    __syncthreads();

    const int pix = blockIdx.x * 256 + threadIdx.x;    // B*H*W = 65536 total
    const int b  = pix / HW_;
    const int hw = pix % HW_;
    const int h  = hw / W_;
    const int w  = hw % W_;

    float acc[27];
    #pragma unroll
    for (int o = 0; o < 27; ++o) acc[o] = ob[o];

    const float* xb = x + (size_t)b * C_ * HW_;
    for (int c = 0; c < C_; ++c) {
        const float* xc = xb + c * HW_;
        #pragma unroll
        for (int ky = 0; ky < 3; ++ky) {
            const int yy = h - 1 + ky;
            if ((unsigned)yy >= (unsigned)H_) continue;
            #pragma unroll
            for (int kx = 0; kx < 3; ++kx) {
                const int xx = w - 1 + kx;
                if ((unsigned)xx >= (unsigned)W_) continue;
                const float xv = xc[yy * W_ + xx];
                const float* wp = s_w + c * 9 + ky * 3 + kx;
                #pragma unroll
                for (int o = 0; o < 27; ++o)
                    acc[o] = fmaf(xv, wp[o * KDIM], acc[o]);
            }
        }
    }

    #pragma unroll
    for (int kk = 0; kk < KK_; ++kk) {
        const int sidx = ((b * KK_ + kk) * H_ + h) * W_ + w;
        dyA[sidx] = acc[kk];
        dxA[sidx] = acc[9 + kk];
        const float z = acc[18 + kk];
        mA[sidx]  = 1.0f / (1.0f + __expf(-z));        // sigmoid
    }
}

// ---------------------------------------------------------------------------
// Kernel 2: fused deformable-im2col + GEMM via V_WMMA_F32_16X16X4_F32.
// One block per (b, h): N-tile = 128 columns (w = 0..127), M = all 64 outputs.
// K = 576 processed in channel-chunks of 4 (36 K-rows staged in LDS).
// 8 waves: wave = (m_tile = wid&3) x (n_half = wid>>2); 4 v8f accumulators.
//
// Col tile is stored K-pair-interleaved: s_col2[kpair][n] = {col[2k][n],
// col[2k+1][n]} so a B fragment is a single aligned ds_load_b64 per lane
// (no VGPR repacking). COLP=144 -> 2*COLP%64 == 32 -> the two half-wave
// row groups hit disjoint LDS banks.
// ---------------------------------------------------------------------------
constexpr int CCH  = 4;            // channels per K-chunk
constexpr int KCH  = CCH * KK_;    // 36 K-rows per chunk
constexpr int KPR  = KCH / 2;      // 18 K-pairs per chunk
constexpr int COLP = 144;          // K-pair row stride (in float2 elements)

__global__ __launch_bounds__(256)
void dcn_main_kernel(const float* __restrict__ x,
                     const float* __restrict__ A,     // dcn_w flat [64][576]
                     const float* __restrict__ bias,  // [64]
                     const float* __restrict__ dyA,
                     const float* __restrict__ dxA,
                     const float* __restrict__ mA,
                     float* __restrict__ out)         // [B][64][H][W]
{
    __shared__ int   s_i00[KK_ * W_], s_i01[KK_ * W_], s_i10[KK_ * W_], s_i11[KK_ * W_];
    __shared__ float s_w00[KK_ * W_], s_w01[KK_ * W_], s_w10[KK_ * W_], s_w11[KK_ * W_];
    __shared__ v2f   s_col2[KPR * COLP];               // 18*144*8 = 20736 B

    const int b = blockIdx.x / H_;
    const int h = blockIdx.x % H_;
    const int tid = threadIdx.x;

    // Per-(kk, w) bilinear coefficients: shared by all 64 channels.
    for (int e = tid; e < KK_ * W_; e += 256) {
        const int kk = e / W_;
        const int w  = e % W_;
        const int sidx = ((b * KK_ + kk) * H_ + h) * W_ + w;
        const float dy = dyA[sidx];
        const float dx = dxA[sidx];
        const float mk = mA[sidx];
        const int ky = kk / 3, kx = kk % 3;
        const float ys = (float)(h - 1 + ky) + dy;
        const float xs = (float)(w - 1 + kx) + dx;
        const float y0f = floorf(ys), x0f = floorf(xs);
        const float ty = ys - y0f, tx = xs - x0f;
        const int y0 = (int)y0f, x0 = (int)x0f;
        const int y1 = y0 + 1,   x1 = x0 + 1;
        const float vy0 = ((unsigned)y0 < (unsigned)H_) ? 1.0f : 0.0f;
        const float vy1 = ((unsigned)y1 < (unsigned)H_) ? 1.0f : 0.0f;
        const float vx0 = ((unsigned)x0 < (unsigned)W_) ? 1.0f : 0.0f;
        const float vx1 = ((unsigned)x1 < (unsigned)W_) ? 1.0f : 0.0f;
        const int cy0 = min(max(y0, 0), H_ - 1);
        const int cy1 = min(max(y1, 0), H_ - 1);
        const int cx0 = min(max(x0, 0), W_ - 1);
        const int cx1 = min(max(x1, 0), W_ - 1);
        s_i00[e] = cy0 * W_ + cx0;
        s_i01[e] = cy0 * W_ + cx1;
        s_i10[e] = cy1 * W_ + cx0;
        s_i11[e] = cy1 * W_ + cx1;
        s_w00[e] = (1.0f - ty) * (1.0f - tx) * mk * vy0 * vx0;
        s_w01[e] = (1.0f - ty) * tx          * mk * vy0 * vx1;
        s_w10[e] = ty * (1.0f - tx)          * mk * vy1 * vx0;
        s_w11[e] = ty * tx                   * mk * vy1 * vx1;
    }
    __syncthreads();

    const int lane  = tid & 31;
    const int wid   = tid >> 5;
    const int mt    = wid & 3;       // M tile: rows mt*16 .. mt*16+15
    const int nh    = wid >> 2;      // N half: cols nh*64 .. nh*64+63
    const int l16   = lane & 15;
    const int ahalf = lane >> 4;     // K-pair selector within a fragment
    const int mrow  = mt * 16 + l16; // A-matrix row held by this lane

    // Accumulators initialized with bias (C matrix), per documented C/D layout:
    // VGPR r: lanes 0-15 -> M = r, lanes 16-31 -> M = r + 8.
    v8f acc[4];
    #pragma unroll
    for (int r = 0; r < 8; ++r) {
        const float bz = bias[mt * 16 + r + (ahalf ? 8 : 0)];
        #pragma unroll
        for (int nt = 0; nt < 4; ++nt) acc[nt][r] = bz;
    }

    const float* xb = x + (size_t)b * C_ * HW_;

    for (int cc = 0; cc < C_ / CCH; ++cc) {
        // Stage the 36 x 128 deformable-sampled col tile into LDS, pairwise:
        // each thread computes both K values of one (kpair, n) -> one b64 store.
        for (int e = tid; e < KPR * W_; e += 256) {
            const int kp = e / W_;
            const int n  = e % W_;
            v2f pr;
            #pragma unroll
            for (int half = 0; half < 2; ++half) {
                const int kl = kp * 2 + half;        // local K row = cl*9 + kk
                const int cl = kl / KK_;
                const int kk = kl % KK_;
                const int c  = cc * CCH + cl;
                const float* xp = xb + c * HW_;
                const int ce = kk * W_ + n;
                pr[half] = s_w00[ce] * xp[s_i00[ce]]
                         + s_w01[ce] * xp[s_i01[ce]]
                         + s_w10[ce] * xp[s_i10[ce]]
                         + s_w11[ce] * xp[s_i11[ce]];
            }
            s_col2[kp * COLP + n] = pr;
        }
        __syncthreads();

        // GEMM over this K-chunk: 9 wmma steps x 4 N tiles.
        const float* Ak = A + (size_t)mrow * KDIM + cc * KCH;
        #pragma unroll
        for (int ks = 0; ks < KCH / 4; ++ks) {
            const int k0 = ks * 4;
            // A fragment (16x4 f32): VGPR0 lanes0-15 K=0 / lanes16-31 K=2; VGPR1: K=1 / K=3
            v2f afrag;
            afrag[0] = Ak[k0 + ahalf * 2 + 0];
            afrag[1] = Ak[k0 + ahalf * 2 + 1];
            const int kpair = ks * 2 + ahalf;        // pair row holding K = k0+2*ahalf, +1
            #pragma unroll
            for (int nt = 0; nt < 4; ++nt) {
                const int n = nh * 64 + nt * 16 + l16;
                const v2f bfrag = s_col2[kpair * COLP + n];  // single ds_load_b64
                acc[nt] = __builtin_amdgcn_wmma_f32_16x16x4_f32(
                    /*neg_a=*/false, afrag, /*neg_b=*/false, bfrag,
                    /*c_mod=*/(short)0, acc[nt],
                    /*reuse_a=*/false, /*reuse_b=*/false);
            }
        }
        __syncthreads();
    }

    // Store D fragments: out[b][row][h][w]
    float* ob = out + (size_t)b * O_ * HW_ + h * W_;
    #pragma unroll
    for (int nt = 0; nt < 4; ++nt) {
        const int n = nh * 64 + nt * 16 + l16;
        #pragma unroll
        for (int r = 0; r < 8; ++r) {
            const int row = mt * 16 + r + (ahalf ? 8 : 0);
            ob[(size_t)row * HW_ + n] = acc[nt][r];
        }
    }
}

// ---------------------------------------------------------------------------
// Host-side launcher
// ---------------------------------------------------------------------------
extern "C" void kernel_launch(void* const* d_in, const int* in_sizes, int n_in,
                              void* d_out, int out_size, void* d_ws, size_t ws_size,
                              hipStream_t stream) {
    const float* x   = (const float*)d_in[0];  // [4][64][128][128]
    const float* ow  = (const float*)d_in[1];  // [27][64][3][3]
    const float* ob  = (const float*)d_in[2];  // [27]
    const float* dw  = (const float*)d_in[3];  // [64][64][3][3] == [64][576]
    const float* db  = (const float*)d_in[4];  // [64]
    float* out = (float*)d_out;                // [4][64][128][128]

    const size_t S = (size_t)B_ * KK_ * HW_;   // 589824 floats per field
    float* dyA = (float*)d_ws;
    float* dxA = dyA + S;
    float* mA  = dxA + S;

    // 1) offset/mask conv: 65536 pixels / 256 threads
    dcn_offset_kernel<<<(B_ * HW_) / 256, 256, 0, stream>>>(x, ow, ob, dyA, dxA, mA);

    // 2) fused deformable sampling + WMMA GEMM: one block per (b, h) row
    dcn_main_kernel<<<B_ * H_, 256, 0, stream>>>(x, dw, db, dyA, dxA, mA, out);
}